// LuxrNet_90125593739372
// MI455X (gfx1250) — compile-verified
//
#include <hip/hip_runtime.h>
#include <hip/hip_bf16.h>

// ---------------------------------------------------------------------------
// MI455X (gfx1250) implementation.
//  * All GEMM-shaped ops use V_WMMA_F32_16X16X4_F32 (exact f32 path; the net
//    is ~16 GFLOP total, far below roofline, so full precision is free).
//  * The inherently-sequential 2x32768-step GRU scan runs as one persistent
//    384-thread workgroup with whh staged in 198KB of LDS (CDNA5: 320KB/WGP),
//    rows padded to 129 floats for conflict-free bank access.
// ---------------------------------------------------------------------------

typedef __attribute__((ext_vector_type(2))) float v2f;
typedef __attribute__((ext_vector_type(8))) float v8f;

// ------------------------- conv1: 10->64, 8x8, s2 ---------------------------
// in  (512,10,30,30)  out (512,64,12,12) with ReLU
__global__ void conv1_k(const float* __restrict__ x, const float* __restrict__ w,
                        const float* __restrict__ b, float* __restrict__ y) {
    long i = (long)blockIdx.x * 256 + threadIdx.x;
    if (i >= 512L * 64 * 144) return;
    int ow = i % 12, oh = (i / 12) % 12, oc = (i / 144) % 64, n = i / (144 * 64);
    const float* xp = x + (long)n * 10 * 900;
    const float* wp = w + (long)oc * 10 * 64;
    float acc = b[oc];
    int ih0 = oh * 2, iw0 = ow * 2;
    for (int ic = 0; ic < 10; ++ic)
        for (int kh = 0; kh < 8; ++kh) {
            const float* xr = xp + ic * 900 + (ih0 + kh) * 30 + iw0;
            const float* wr = wp + ic * 64 + kh * 8;
#pragma unroll
            for (int kw = 0; kw < 8; ++kw) acc += xr[kw] * wr[kw];
        }
    y[i] = fmaxf(acc, 0.f);
}

// ------------------------- conv2: 64->128, 4x4, s2 --------------------------
// in (512,64,12,12)  out stored flattened (512, 3200) = [n][oc*25+oh*5+ow], ReLU
__global__ void conv2_k(const float* __restrict__ x, const float* __restrict__ w,
                        const float* __restrict__ b, float* __restrict__ y) {
    long i = (long)blockIdx.x * 256 + threadIdx.x;
    if (i >= 512L * 128 * 25) return;
    int ow = i % 5, oh = (i / 5) % 5, oc = (i / 25) % 128, n = i / 3200;
    float acc = b[oc];
    int ih0 = oh * 2, iw0 = ow * 2;
    for (int ic = 0; ic < 64; ++ic) {
        const float* xr = x + (((long)n * 64 + ic) * 12 + ih0) * 12 + iw0;
        const float* wr = w + (((long)oc * 64 + ic) * 4) * 4;
#pragma unroll
        for (int kh = 0; kh < 4; ++kh)
#pragma unroll
            for (int kw = 0; kw < 4; ++kw)
                acc += xr[kh * 12 + kw] * wr[kh * 4 + kw];
    }
    y[(long)n * 3200 + oc * 25 + oh * 5 + ow] = fmaxf(acc, 0.f);
}

// ----------------- seq = transpose(units,(1,0,2)).reshape(T*B,8) ------------
__global__ void seq_k(const float* __restrict__ u, float* __restrict__ s) {
    int i = blockIdx.x * 256 + threadIdx.x;  // 32768*8 elements
    if (i >= 32768 * 8) return;
    int c = i & 7;
    int row = i >> 3;            // t*512 + b
    int t = row >> 9, bb = row & 511;
    s[i] = u[((long)bb * 64 + t) * 8 + c];
}

// --------------------- generic WMMA f32 GEMM: C = A @ W^T + bias ------------
// A: M x K (lda), W: N x K (ldw), C: M x N region inside row-stride ldc.
// One wave32 computes one 16x16 tile with V_WMMA_F32_16X16X4_F32.
// Per-lane fragment layouts per CDNA5 ISA 7.12.2:
//   A 16x4 : lane L -> row M=L%16; VGPR0=K(2*(L/16)), VGPR1=K(2*(L/16)+1)
//   B 4x16 : lane L -> col N=L%16; VGPR0=K(2*(L/16)), VGPR1=K(2*(L/16)+1)
//   C 16x16: lane L -> col N=L%16; acc[i] = row (i + 8*(L/16))
__global__ void gemm_wmma_k(const float* __restrict__ A, int lda,
                            const float* __restrict__ W, int ldw,
                            const float* __restrict__ bias,
                            float* __restrict__ C, int ldc,
                            int M, int N, int K, int relu) {
    int wave = blockIdx.x * (blockDim.x >> 5) + (threadIdx.x >> 5);
    int lane = threadIdx.x & 31;
    int tilesN = N >> 4;
    int tilesTotal = (M >> 4) * tilesN;
    if (wave >= tilesTotal) return;          // wave-uniform: EXEC stays all-1s for WMMA
    int tm = wave / tilesN, tn = wave % tilesN;

    int idx16 = lane & 15;                   // A: row in tile; B: col in tile
    int khalf = lane >> 4;                   // 0 or 1 -> K pair 0..1 / 2..3
    const float* Arow = A + (long)(tm * 16 + idx16) * lda + 2 * khalf;
    const float* Wrow = W + (long)(tn * 16 + idx16) * ldw + 2 * khalf;

    v8f acc = {0.f, 0.f, 0.f, 0.f, 0.f, 0.f, 0.f, 0.f};
    for (int k0 = 0; k0 < K; k0 += 4) {
        v2f a, bfrag;
        a.x     = Arow[k0];
        a.y     = Arow[k0 + 1];
        bfrag.x = Wrow[k0];
        bfrag.y = Wrow[k0 + 1];
        acc = __builtin_amdgcn_wmma_f32_16x16x4_f32(
            /*neg_a=*/false, a, /*neg_b=*/false, bfrag,
            /*c_mod=*/(short)0, acc, /*reuse_a=*/false, /*reuse_b=*/false);
    }

    int col = tn * 16 + idx16;
    int rbase = tm * 16 + (khalf << 3);
    float bcol = bias ? bias[col] : 0.f;
#pragma unroll
    for (int i = 0; i < 8; ++i) {
        float v = acc[i] + bcol;
        if (relu) v = fmaxf(v, 0.f);
        C[(long)(rbase + i) * ldc + col] = v;
    }
}

// ----------------------------- sequential GRU scan --------------------------
// One persistent 384-thread workgroup. whh (384x128) staged in LDS with rows
// padded to 129 floats (bank-conflict-free). h (128) and gh (384) in LDS.
// Dynamic LDS = (384*129 + 128 + 384)*4 = 200192 B (CDNA5 allows 320KB/WGP).
__global__ void gru_scan_k(const float* __restrict__ gi,   // steps x 384
                           const float* __restrict__ whh,  // 384 x 128
                           const float* __restrict__ bhh,  // 384
                           float* __restrict__ out,        // see ldc/coloff
                           int steps, int out_ldc, int out_coloff, int out_last) {
    extern __shared__ float smem[];
    float* whh_s = smem;                 // 384 * 129
    float* h     = smem + 384 * 129;     // 128
    float* gh    = h + 128;              // 384
    const int tid = threadIdx.x;         // 0..383, one gate row per thread

    // stage whh into LDS (padded rows)
    for (int k = 0; k < 128; ++k) whh_s[tid * 129 + k] = whh[tid * 128 + k];
    if (tid < 128) h[tid] = 0.f;
    __syncthreads();

    const float bh = bhh[tid];
    const int first = steps - out_last;
    for (int s = 0; s < steps; ++s) {
        if (s + 1 < steps) __builtin_prefetch(gi + (long)(s + 1) * 384, 0, 0);
        // gh = whh @ h + bhh  (128-MAC LDS dot per lane)
        float acc = bh;
        const float* wr = &whh_s[tid * 129];
#pragma unroll 8
        for (int k = 0; k < 128; ++k) acc += wr[k] * h[k];
        gh[tid] = acc;
        __syncthreads();
        if (tid < 128) {
            const float* g = gi + (long)s * 384;
            float r  = 1.f / (1.f + __expf(-(g[tid]       + gh[tid])));
            float z  = 1.f / (1.f + __expf(-(g[128 + tid] + gh[128 + tid])));
            float nn = tanhf(g[256 + tid] + r * gh[256 + tid]);
            float hn = (1.f - z) * nn + z * h[tid];
            h[tid] = hn;
            if (s >= first)
                out[(long)(s - first) * out_ldc + out_coloff + tid] = hn;
        }
        __syncthreads();
    }
}

// ------------------------- final: out = hidden @ out_w^T + b ----------------
__global__ void out_k(const float* __restrict__ hid, const float* __restrict__ w,
                      const float* __restrict__ b, float* __restrict__ o) {
    int i = blockIdx.x * 256 + threadIdx.x;  // 512*12
    if (i >= 512 * 12) return;
    int oo = i % 12, n = i / 12;
    float acc = b[oo];
    const float* hr = hid + (long)n * 512;
    const float* wr = w + (long)oo * 512;
#pragma unroll 8
    for (int k = 0; k < 512; ++k) acc += hr[k] * wr[k];
    o[i] = acc;
}

// ---------------------------------------------------------------------------
static inline void launch_gemm(const float* A, int lda, const float* W, int ldw,
                               const float* bias, float* C, int ldc,
                               int M, int N, int K, int relu, hipStream_t s) {
    int tiles = (M >> 4) * (N >> 4);
    int blocks = (tiles + 7) / 8;  // 8 wave32 per 256-thread block
    gemm_wmma_k<<<blocks, 256, 0, s>>>(A, lda, W, ldw, bias, C, ldc, M, N, K, relu);
}

extern "C" void kernel_launch(void* const* d_in, const int* in_sizes, int n_in,
                              void* d_out, int out_size, void* d_ws, size_t ws_size,
                              hipStream_t stream) {
    const float* gfeat   = (const float*)d_in[0];
    const float* units   = (const float*)d_in[1];
    const float* conv1_w = (const float*)d_in[2];
    const float* conv1_b = (const float*)d_in[3];
    const float* conv2_w = (const float*)d_in[4];
    const float* conv2_b = (const float*)d_in[5];
    const float* ffn_w   = (const float*)d_in[6];
    const float* ffn_b   = (const float*)d_in[7];
    const float* wih0    = (const float*)d_in[8];
    const float* whh0    = (const float*)d_in[9];
    const float* bih0    = (const float*)d_in[10];
    const float* bhh0    = (const float*)d_in[11];
    const float* wih1    = (const float*)d_in[12];
    const float* whh1    = (const float*)d_in[13];
    const float* bih1    = (const float*)d_in[14];
    const float* bhh1    = (const float*)d_in[15];
    const float* joiner_w = (const float*)d_in[16];
    const float* joiner_b = (const float*)d_in[17];
    const float* out_w   = (const float*)d_in[18];
    const float* out_b   = (const float*)d_in[19];
    float* out = (float*)d_out;

    // workspace layout (floats); ~79MB total -> fully L2-resident on MI455X
    float* ws  = (float*)d_ws;
    float* A1  = ws;                   // conv1 out: 512*64*144   = 4,718,592
    float* A2  = A1 + 4718592;         // conv2 out: 512*3200     = 1,638,400
    float* seq = A2 + 1638400;         // 32768*8                 =   262,144
    float* gi  = seq + 262144;         // 32768*384               = 12,582,912
    float* cc  = gi + 12582912;        // concat: 512*640         =   327,680
    float* hid = cc + 327680;          // 512*512                 =   262,144
    float* h1  = A1;                   // reuse conv1 buffer after conv2 done

    const int SCAN_SMEM = (384 * 129 + 128 + 384) * 4;  // 200,192 B LDS

    // 1-2: convolutions (direct; compute-trivial, L2-resident)
    conv1_k<<<18432, 256, 0, stream>>>(gfeat, conv1_w, conv1_b, A1);
    conv2_k<<<6400, 256, 0, stream>>>(A1, conv2_w, conv2_b, A2);

    // 3: build GRU input sequence
    seq_k<<<1024, 256, 0, stream>>>(units, seq);

    // 4: gi0 = seq @ wih0^T + bih0            (32768 x 384, K=8)
    launch_gemm(seq, 8, wih0, 8, bih0, gi, 384, 32768, 384, 8, 0, stream);

    // 5: glob = conv_flat @ ffn^T + b  -> concat cols [0,512)  (K=3200)
    launch_gemm(A2, 3200, ffn_w, 3200, ffn_b, cc, 640, 512, 512, 3200, 0, stream);

    // 6: GRU layer 0 scan (sequential, 32768 steps) -> h1
    gru_scan_k<<<1, 384, SCAN_SMEM, stream>>>(gi, whh0, bhh0, h1, 32768, 128, 0, 32768);

    // 7: gi1 = h1 @ wih1^T + bih1             (32768 x 384, K=128), overwrite gi
    launch_gemm(h1, 128, wih1, 128, bih1, gi, 384, 32768, 384, 128, 0, stream);

    // 8: GRU layer 1 scan; keep only last 512 states -> concat cols [512,640)
    gru_scan_k<<<1, 384, SCAN_SMEM, stream>>>(gi, whh1, bhh1, cc, 32768, 640, 512, 512);

    // 9: hidden = relu(concat @ joiner^T + b)  (512 x 512, K=640)
    launch_gemm(cc, 640, joiner_w, 640, joiner_b, hid, 512, 512, 512, 640, 1, stream);

    // 10: out = hidden @ out_w^T + out_b       (512 x 12)
    out_k<<<24, 256, 0, stream>>>(hid, out_w, out_b, out);
}